// Topk_75788992905345
// MI455X (gfx1250) — compile-verified
//
#include <hip/hip_runtime.h>

// ---------------------------------------------------------------------------
// Top-k masking (k = 0.1*N) via 2-level radix select.
//   K1: 12-bit LDS histogram of monotone keys, streamed through LDS with
//       async global->LDS loads (ASYNCcnt double-buffer pipeline)
//   K2: scan 4096 bins -> threshold bucket B, remaining count r
//   K3: collect (key,idx) candidates in bucket B (1 streaming read)
//   K4: exact radix-select on low 20 bits -> T, tEq, Istar (tiny)
//   K5: out = keep ? x : 0 (1 read + 1 write)
// Deterministic: tie resolution re-derived from values/indices, not from
// atomic arrival order.
// ---------------------------------------------------------------------------

#define HIST_BITS 12
#define HIST_BINS (1 << HIST_BITS)
#define KEY_SHIFT (32 - HIST_BITS)          // 20
#define CAND_CAP  65536u                    // expected ~N/4096 ~ 8192
#define HBLK      256                       // K1 block size (8 wave32)

// params layout (u32): 0=B  1=r  2=T  3=tEq  4=Istar  5=candCount
#define P_B     0
#define P_R     1
#define P_T     2
#define P_TEQ   3
#define P_ISTAR 4
#define P_CNT   5

typedef int v4i __attribute__((ext_vector_type(4)));
typedef __attribute__((address_space(1))) v4i global_v4i;
typedef __attribute__((address_space(3))) v4i shared_v4i;

__device__ __forceinline__ unsigned mono_key(float f) {
    // order-preserving bijection float -> u32 (no NaNs in input)
    unsigned b = __float_as_uint(f);
    unsigned m = (unsigned)(((int)b) >> 31) | 0x80000000u;
    return b ^ m;
}

// ----- CDNA5 async global->LDS (ASYNCcnt) helpers --------------------------
__device__ __forceinline__ void async_stage_b128(const float4* g, float4* l) {
#if __has_builtin(__builtin_amdgcn_global_load_async_to_lds_b128)
    __builtin_amdgcn_global_load_async_to_lds_b128(
        (global_v4i*)g,        // addrspace(1) v4i*
        (shared_v4i*)l,        // addrspace(3) v4i*
        /*offset=*/0, /*cpol=*/0);
#else
    // VDST = LDS byte address (low 32 bits of flat shared pointer),
    // VADDR = 64-bit global address.
    unsigned lptr = (unsigned)(unsigned long long)l;
    asm volatile("global_load_async_to_lds_b128 %0, %1, off"
                 :: "v"(lptr), "v"(g)
                 : "memory");
#endif
}

__device__ __forceinline__ void wait_async_le1() {
#if __has_builtin(__builtin_amdgcn_s_wait_asynccnt)
    __builtin_amdgcn_s_wait_asynccnt(1);
#else
    asm volatile("s_wait_asynccnt 0x1" ::: "memory");
#endif
}

__device__ __forceinline__ void wait_async_le0() {
#if __has_builtin(__builtin_amdgcn_s_wait_asynccnt)
    __builtin_amdgcn_s_wait_asynccnt(0);
#else
    asm volatile("s_wait_asynccnt 0x0" ::: "memory");
#endif
}

// ---------------------------------------------------------------- K1: hist
// Double-buffered async global->LDS staging; each wave consumes only its own
// staged slots, so no cross-wave sync is needed inside the pipeline loop.
__global__ void __launch_bounds__(HBLK) k_hist(const float4* __restrict__ x4,
                                               int n4,
                                               unsigned* __restrict__ ghist) {
    __shared__ unsigned lh[HIST_BINS];
    __shared__ float4 stage[2][HBLK];

    for (int i = threadIdx.x; i < HIST_BINS; i += blockDim.x) lh[i] = 0u;
    __syncthreads();

    const int tid    = threadIdx.x;
    const int stride = (int)(gridDim.x * blockDim.x);
    const int i0     = blockIdx.x * blockDim.x + tid;

    int buf = 0;
    if (i0 < n4) async_stage_b128(&x4[i0], &stage[0][tid]);

    for (int i = i0; i < n4; i += stride) {
        const int inext = i + stride;
        if (inext < n4) {
            async_stage_b128(&x4[inext], &stage[buf ^ 1][tid]);  // prefetch tile t+1
            wait_async_le1();                                    // tile t landed
        } else {
            wait_async_le0();
        }
        float4 v = stage[buf][tid];                              // ds_load_b128
        atomicAdd(&lh[mono_key(v.x) >> KEY_SHIFT], 1u);
        atomicAdd(&lh[mono_key(v.y) >> KEY_SHIFT], 1u);
        atomicAdd(&lh[mono_key(v.z) >> KEY_SHIFT], 1u);
        atomicAdd(&lh[mono_key(v.w) >> KEY_SHIFT], 1u);
        buf ^= 1;
    }
    __syncthreads();

    for (int i = threadIdx.x; i < HIST_BINS; i += blockDim.x) {
        unsigned c = lh[i];
        if (c) atomicAdd(&ghist[i], c);
    }
}

// ---------------------------------------------------------------- K2: scan
__global__ void __launch_bounds__(1024) k_scan_hi(const unsigned* __restrict__ ghist,
                                                  unsigned* __restrict__ params,
                                                  unsigned K) {
    __shared__ unsigned sh[HIST_BINS];
    for (int i = threadIdx.x; i < HIST_BINS; i += blockDim.x) sh[i] = ghist[i];
    __syncthreads();
    if (threadIdx.x == 0) {
        unsigned acc = 0u, B = 0u, r = K;
        for (int b = HIST_BINS - 1; b >= 0; --b) {
            unsigned c = sh[b];
            if (acc + c >= K) { B = (unsigned)b; r = K - acc; break; }
            acc += c;
        }
        params[P_B] = B;
        params[P_R] = r;
    }
}

// ------------------------------------------------------------- K3: collect
__global__ void __launch_bounds__(256) k_collect(const float4* __restrict__ x4,
                                                 int n4,
                                                 unsigned* __restrict__ params,
                                                 unsigned* __restrict__ ckey,
                                                 unsigned* __restrict__ cidx) {
    const unsigned B = params[P_B];
    const int stride = (int)(gridDim.x * blockDim.x);
    for (int i = blockIdx.x * blockDim.x + threadIdx.x; i < n4; i += stride) {
        __builtin_prefetch(&x4[i + stride], 0, 0);   // global_prefetch_b8
        float4 v = x4[i];
        unsigned base = 4u * (unsigned)i;
        unsigned kk;

        kk = mono_key(v.x);
        if ((kk >> KEY_SHIFT) == B) {
            unsigned p = atomicAdd(&params[P_CNT], 1u);
            if (p < CAND_CAP) { ckey[p] = kk; cidx[p] = base + 0u; }
        }
        kk = mono_key(v.y);
        if ((kk >> KEY_SHIFT) == B) {
            unsigned p = atomicAdd(&params[P_CNT], 1u);
            if (p < CAND_CAP) { ckey[p] = kk; cidx[p] = base + 1u; }
        }
        kk = mono_key(v.z);
        if ((kk >> KEY_SHIFT) == B) {
            unsigned p = atomicAdd(&params[P_CNT], 1u);
            if (p < CAND_CAP) { ckey[p] = kk; cidx[p] = base + 2u; }
        }
        kk = mono_key(v.w);
        if ((kk >> KEY_SHIFT) == B) {
            unsigned p = atomicAdd(&params[P_CNT], 1u);
            if (p < CAND_CAP) { ckey[p] = kk; cidx[p] = base + 3u; }
        }
    }
}

// -------------------------------------------------------------- K4: select
__global__ void __launch_bounds__(1024) k_select(unsigned* __restrict__ params,
                                                 const unsigned* __restrict__ ckey,
                                                 const unsigned* __restrict__ cidx) {
    __shared__ unsigned red[1024];

    unsigned cnt = params[P_CNT];
    if (cnt > CAND_CAP) cnt = CAND_CAP;
    unsigned need   = params[P_R];               // >= 1
    unsigned prefix = params[P_B] << KEY_SHIFT;  // decided high bits

    // exact radix select over low KEY_SHIFT bits
    for (int bit = KEY_SHIFT - 1; bit >= 0; --bit) {
        const unsigned bm = 1u << bit;
        const unsigned hm = ~((bm << 1) - 1u);   // mask of bits [31:bit+1]
        unsigned local = 0u;
        for (unsigned i = threadIdx.x; i < cnt; i += blockDim.x) {
            unsigned kk = ckey[i];
            if (((kk & hm) == prefix) && (kk & bm)) local++;
        }
        red[threadIdx.x] = local;
        __syncthreads();
        for (int s = (int)blockDim.x >> 1; s > 0; s >>= 1) {
            if ((int)threadIdx.x < s) red[threadIdx.x] += red[threadIdx.x + s];
            __syncthreads();
        }
        unsigned c1 = red[0];
        __syncthreads();
        if (c1 >= need) prefix |= bm;            // threshold has this bit set
        else            need  -= c1;             // all c1 elements are kept
    }

    if (threadIdx.x == 0) {
        params[P_T]   = prefix;                  // exact 32-bit threshold key
        params[P_TEQ] = need;                    // #ties to keep (>=1)
        // Istar = need-th smallest flat index among exact ties (lowest-index
        // tie-break, matching jax.lax.top_k). Tie population is tiny.
        unsigned prev = 0u;
        bool first = true;
        for (unsigned t = 0; t < need; ++t) {
            unsigned best = 0xFFFFFFFFu;
            for (unsigned i = 0; i < cnt; ++i) {
                if (ckey[i] == prefix) {
                    unsigned id = cidx[i];
                    if ((first || id > prev) && id < best) best = id;
                }
            }
            prev = best;
            first = false;
        }
        params[P_ISTAR] = prev;
    }
}

// --------------------------------------------------------------- K5: final
__device__ __forceinline__ float sel_one(float f, unsigned idx,
                                         unsigned T, unsigned tEq, unsigned Istar) {
    unsigned kk = mono_key(f);
    bool keep = (kk > T) || ((kk == T) && (tEq != 0u) && (idx <= Istar));
    return keep ? f : 0.0f;
}

__global__ void __launch_bounds__(256) k_final(const float4* __restrict__ x4,
                                               float4* __restrict__ out4,
                                               int n4,
                                               const unsigned* __restrict__ params) {
    const unsigned T     = params[P_T];
    const unsigned tEq   = params[P_TEQ];
    const unsigned Istar = params[P_ISTAR];
    const int stride = (int)(gridDim.x * blockDim.x);
    for (int i = blockIdx.x * blockDim.x + threadIdx.x; i < n4; i += stride) {
        __builtin_prefetch(&x4[i + stride], 0, 0);
        float4 v = x4[i];
        unsigned base = 4u * (unsigned)i;
        float4 o;
        o.x = sel_one(v.x, base + 0u, T, tEq, Istar);
        o.y = sel_one(v.y, base + 1u, T, tEq, Istar);
        o.z = sel_one(v.z, base + 2u, T, tEq, Istar);
        o.w = sel_one(v.w, base + 3u, T, tEq, Istar);
        out4[i] = o;
    }
}

// ---------------------------------------------------------------------------
extern "C" void kernel_launch(void* const* d_in, const int* in_sizes, int n_in,
                              void* d_out, int out_size, void* d_ws, size_t ws_size,
                              hipStream_t stream) {
    const float* x = (const float*)d_in[0];
    float* out = (float*)d_out;
    const int n  = in_sizes[0];          // 8*1024*4096 = 33,554,432 (mult. of 4)
    const int n4 = n >> 2;
    const unsigned K = (unsigned)(0.1 * (double)n);   // matches Python int(RATIO*N)

    unsigned char* ws = (unsigned char*)d_ws;
    unsigned* ghist  = (unsigned*)(ws);                                     // 16 KB
    unsigned* params = (unsigned*)(ws + HIST_BINS * 4);                     // 64 B
    unsigned* ckey   = (unsigned*)(ws + HIST_BINS * 4 + 64);                // 256 KB
    unsigned* cidx   = (unsigned*)(ws + HIST_BINS * 4 + 64 + CAND_CAP * 4); // 256 KB

    // zero hist + params (incl. candidate counter) every launch — the harness
    // does not re-poison between graph replays.
    (void)hipMemsetAsync(d_ws, 0, HIST_BINS * 4 + 64, stream);

    const int blocks = 2048;   // n4/(2048*256) = 16 iterations, no tail divergence

    k_hist   <<<blocks, HBLK, 0, stream>>>((const float4*)x, n4, ghist);
    k_scan_hi<<<1, 1024, 0, stream>>>(ghist, params, K);
    k_collect<<<blocks, 256, 0, stream>>>((const float4*)x, n4, params, ckey, cidx);
    k_select <<<1, 1024, 0, stream>>>(params, ckey, cidx);
    k_final  <<<blocks, 256, 0, stream>>>((const float4*)x, (float4*)out, n4, params);
}